// PersistentAssoMemBlock_14534169329972
// MI455X (gfx1250) — compile-verified
//
#include <hip/hip_runtime.h>
#include <math.h>
#include <stdint.h>

typedef __attribute__((ext_vector_type(2))) float v2f;
typedef __attribute__((ext_vector_type(8))) float v8f;

#define PADS 68   // LDS row stride (floats) for staged K/V tiles: bank-conflict free, 8B aligned

// ---------------------------------------------------------------------------
// FP32 WMMA: D(16x16) = A(16x4) * B(4x16) + C   (wave32, gfx1250)
// A frag: lane m=lane&15 holds {A[m][2*(lane>>4)], A[m][2*(lane>>4)+1]}
// B frag: lane n=lane&15 holds {B[2*(lane>>4)][n], B[2*(lane>>4)+1][n]}
// C/D:    VGPR r: lanes 0-15 -> (M=r, N=lane); lanes 16-31 -> (M=r+8, N=lane-16)
// ---------------------------------------------------------------------------
__device__ __forceinline__ v8f wmma_f32_16x16x4(v2f a, v2f b, v8f c) {
  return __builtin_amdgcn_wmma_f32_16x16x4_f32(
      false, a, false, b, (short)0, c, false, false);
}

// ---------------------------------------------------------------------------
// Y[i][j] = sum_k X[i][k] * W[j][k]   (X: Nr x K, W: Nc x K, Y: Nr x Nc)
// One wave per 16x64 stripe.  K-loop steps by 16: all 20 fragments are loaded
// first (load clause, progressive loadcnt waits), then 16 WMMAs issue.
// ---------------------------------------------------------------------------
__global__ void gemm_nt_wmma(const float* __restrict__ X,
                             const float* __restrict__ W,
                             float* __restrict__ Y,
                             int Nr, int K, int Nc) {
  const int lane  = threadIdx.x;
  const int m16   = lane & 15;
  const int khalf = (lane >> 4) << 1;          // 0 or 2
  const int tm  = blockIdx.x * 16;
  const int tn0 = (blockIdx.y * blockDim.y + threadIdx.y) * 64;
  if (tm >= Nr || tn0 >= Nc) return;           // wave-uniform predicate

  const float* xrow = X + (size_t)(tm + m16) * K + khalf;
  const float* w0   = W + (size_t)(tn0 + m16) * K + khalf;

  v8f c0 = {}, c1 = {}, c2 = {}, c3 = {};
  for (int k0 = 0; k0 < K; k0 += 16) {
    v2f a[4], b0[4], b1[4], b2[4], b3[4];
#pragma unroll
    for (int j = 0; j < 4; ++j) {
      const int kq = k0 + 4 * j;
      a[j]  = *(const v2f*)(xrow + kq);
      b0[j] = *(const v2f*)(w0 + kq);
      b1[j] = *(const v2f*)(w0 + (size_t)16 * K + kq);
      b2[j] = *(const v2f*)(w0 + (size_t)32 * K + kq);
      b3[j] = *(const v2f*)(w0 + (size_t)48 * K + kq);
    }
#pragma unroll
    for (int j = 0; j < 4; ++j) {
      c0 = wmma_f32_16x16x4(a[j], b0[j], c0);
      c1 = wmma_f32_16x16x4(a[j], b1[j], c1);
      c2 = wmma_f32_16x16x4(a[j], b2[j], c2);
      c3 = wmma_f32_16x16x4(a[j], b3[j], c3);
    }
  }

  const int row0 = (lane < 16) ? 0 : 8;
#pragma unroll
  for (int r = 0; r < 8; ++r) {
    float* yp = Y + (size_t)(tm + row0 + r) * Nc + tn0 + m16;
    yp[0]  = c0[r];
    yp[16] = c1[r];
    yp[32] = c2[r];
    yp[48] = c3[r];
  }
}

// ---------------------------------------------------------------------------
// Leaky EMA over time.  u:(B,T,C) -> kb:(B,NH,T,HS).  One thread per channel;
// at each t the B*C threads read a contiguous row of u (coalesced).
// ---------------------------------------------------------------------------
__global__ void ema_scan(const float* __restrict__ U,
                         const float* __restrict__ leak,
                         float* __restrict__ Kb,
                         int Bn, int Tn, int Cn, int NHn, int HSn) {
  int tid = blockIdx.x * blockDim.x + threadIdx.x;
  if (tid >= Bn * Cn) return;
  int b = tid / Cn, ch = tid % Cn;
  int hh = ch / HSn, d = ch % HSn;
  float alpha = 1.0f / (1.0f + __expf(-leak[hh]));
  float om    = 1.0f - alpha;
  float hstate = 0.0f;
  const float* up = U + (size_t)b * Tn * Cn + ch;
  float* kp = Kb + (((size_t)b * NHn + hh) * Tn) * HSn + d;
  for (int t = 0; t < Tn; ++t) {
    hstate = alpha * hstate + om * up[(size_t)t * Cn];
    kp[(size_t)t * HSn] = hstate;
  }
}

// ---------------------------------------------------------------------------
// Flash-style attention over persistent memory slots.
// Block = 4 waves on 4 consecutive t-tiles of the SAME (b,h).  K/V chunks are
// staged into double-buffered LDS via async global->LDS copies: the next
// chunk's DMA (512B per wave) overlaps the current chunk's 32 WMMAs.
// Query L2-norm + kernel scale folded into the A-fragment preload.
// Online softmax; P transposed through LDS.
// ---------------------------------------------------------------------------
__global__ void pmem_attn(const float* __restrict__ Kb,
                          const float* __restrict__ Kmem,
                          const float* __restrict__ Vmem,
                          const float* __restrict__ kbeta,
                          const float* __restrict__ vbeta,
                          float* __restrict__ Y,
                          int Bn, int Tn, int NHn, int Mn, int HSn) {
  __shared__ float sK[2][16 * PADS];
  __shared__ float sV[2][16 * PADS];
  __shared__ float sP[4][16 * 17];

  const int lane  = threadIdx.x;
  const int m16   = lane & 15;
  const int khalf = (lane >> 4) << 1;
  const int row0  = (lane < 16) ? 0 : 8;

  const int tile = blockIdx.x * blockDim.y + threadIdx.y;
  const int tilesPerHead = Tn >> 4;            // 128, divisible by 4 -> same (b,h) per block
  const int t0 = (tile % tilesPerHead) << 4;
  const int bh = tile / tilesPerHead;
  const int hh = bh % NHn;
  const int b  = bh / NHn;

  // --- cooperative staging assignment: 128 threads move 2KB (K then V) ---
  const int tid  = threadIdx.y * 32 + threadIdx.x;   // 0..127
  const int seg  = tid & 63;                         // 64 x 16B segments per tile
  const int srow = seg >> 2;
  const int scol = (seg & 3) * 4;                    // in floats
  const bool doV = tid >= 64;
  float* stageDst0 = (doV ? sV[0] : sK[0]) + srow * PADS + scol;
  const unsigned ldsOff0   = (unsigned)(uintptr_t)stageDst0;  // LDS byte offset
  const unsigned bufStride = (unsigned)(16 * PADS * sizeof(float));

  const float* Kh = Kmem + (size_t)hh * Mn * HSn;
  const float* Vh = Vmem + (size_t)hh * Mn * HSn;
  const float* stageSrcBase = (doV ? Vh : Kh) + (size_t)srow * HSn + scol;

  // --- preload query A-fragments; fold L2-normalize + exp-kernel scale ---
  const float* qbase = Kb + (((size_t)b * NHn + hh) * Tn + t0) * HSn;
  v2f afr[16];
  float ssq = 0.0f;
#pragma unroll
  for (int kk = 0; kk < 16; ++kk) {
    afr[kk] = *(const v2f*)(qbase + (size_t)m16 * HSn + kk * 4 + khalf);
    ssq += afr[kk].x * afr[kk].x + afr[kk].y * afr[kk].y;
  }
  ssq += __shfl_xor(ssq, 16, 32);                    // combine both k-halves of row m16
  float bbk = fminf(kbeta[hh] * 10.0f, 5.0f);
  float sk2 = __expf(bbk);
  sk2 *= sk2;
  const float qmul = sk2 * rsqrtf(ssq);
#pragma unroll
  for (int kk = 0; kk < 16; ++kk) afr[kk] *= qmul;

  v8f acc0 = {}, acc1 = {}, acc2 = {}, acc3 = {};
  float m_run[8], l_run[8];
#pragma unroll
  for (int r = 0; r < 8; ++r) { m_run[r] = -3.0e38f; l_run[r] = 0.0f; }

  float* myP = &sP[threadIdx.y][0];
  const int nChunks = Mn >> 4;

  // --- prologue: async-stage chunk 0 into buffer 0 ---
  asm volatile("global_load_async_to_lds_b128 %0, %1, off"
               :: "v"(ldsOff0), "v"(stageSrcBase) : "memory");

  for (int ci = 0; ci < nChunks; ++ci) {
    const int cur = ci & 1;
    // issue next chunk's DMA into the other buffer (dummy chunk-0 load on tail)
    const int m0n = (ci + 1 < nChunks) ? (ci + 1) * 16 : 0;
    const float* gnext = stageSrcBase + (size_t)m0n * HSn;
    const unsigned ldsN = ldsOff0 + (unsigned)(cur ^ 1) * bufStride;
    asm volatile("global_load_async_to_lds_b128 %0, %1, off"
                 :: "v"(ldsN), "v"(gnext) : "memory");
    // async ops complete in order: <=1 outstanding means current chunk landed
    asm volatile("s_wait_asynccnt 0x1" ::: "memory");
    __syncthreads();

    const float* cK = sK[cur];
    const float* cV = sV[cur];

    // ---- scores S(16x16) = Q(16x64) * Kchunk^T : batch LDS loads, then WMMAs
    v2f bfr[16];
#pragma unroll
    for (int kk = 0; kk < 16; ++kk)
      bfr[kk] = *(const v2f*)(cK + m16 * PADS + kk * 4 + khalf);
    v8f s = {};
#pragma unroll
    for (int kk = 0; kk < 16; ++kk)
      s = wmma_f32_16x16x4(afr[kk], bfr[kk], s);

    // ---- online softmax (row reductions within each 16-lane half) ----
#pragma unroll
    for (int r = 0; r < 8; ++r) {
      float sv = s[r];
      float rm = sv;
      rm = fmaxf(rm, __shfl_xor(rm, 1, 32));
      rm = fmaxf(rm, __shfl_xor(rm, 2, 32));
      rm = fmaxf(rm, __shfl_xor(rm, 4, 32));
      rm = fmaxf(rm, __shfl_xor(rm, 8, 32));
      float newm = fmaxf(m_run[r], rm);
      float corr = __expf(m_run[r] - newm);
      float p    = __expf(sv - newm);
      float rs   = p;
      rs += __shfl_xor(rs, 1, 32);
      rs += __shfl_xor(rs, 2, 32);
      rs += __shfl_xor(rs, 4, 32);
      rs += __shfl_xor(rs, 8, 32);
      l_run[r] = l_run[r] * corr + rs;
      m_run[r] = newm;
      acc0[r] *= corr; acc1[r] *= corr; acc2[r] *= corr; acc3[r] *= corr;
      s[r] = p;
    }

    // ---- transpose P: D-layout regs -> row-major LDS -> A-layout regs ----
#pragma unroll
    for (int r = 0; r < 8; ++r)
      myP[(row0 + r) * 17 + m16] = s[r];
    asm volatile("s_wait_dscnt 0x0" ::: "memory");

    // ---- acc(16x64) += P(16x16) * Vchunk(16x64): batch LDS loads, then WMMAs
    v2f a2[4], bv0[4], bv1[4], bv2[4], bv3[4];
#pragma unroll
    for (int j = 0; j < 4; ++j) {
      const int kr = j * 4 + khalf;
      a2[j].x = myP[m16 * 17 + kr];
      a2[j].y = myP[m16 * 17 + kr + 1];
      const float* vr0 = cV + kr * PADS + m16;
      bv0[j].x = vr0[0];    bv0[j].y = vr0[PADS];
      bv1[j].x = vr0[16];   bv1[j].y = vr0[PADS + 16];
      bv2[j].x = vr0[32];   bv2[j].y = vr0[PADS + 32];
      bv3[j].x = vr0[48];   bv3[j].y = vr0[PADS + 48];
    }
#pragma unroll
    for (int j = 0; j < 4; ++j) {
      acc0 = wmma_f32_16x16x4(a2[j], bv0[j], acc0);
      acc1 = wmma_f32_16x16x4(a2[j], bv1[j], acc1);
      acc2 = wmma_f32_16x16x4(a2[j], bv2[j], acc2);
      acc3 = wmma_f32_16x16x4(a2[j], bv3[j], acc3);
    }
    __syncthreads();   // all waves done with this buffer before it is re-filled
  }

  // ---- finalize: y = (acc / l) * clip(exp(value_beta*10), 5), merge heads ----
  const float svg = fminf(__expf(vbeta[hh] * 10.0f), 5.0f);
  const int Cn = NHn * HSn;
#pragma unroll
  for (int r = 0; r < 8; ++r) {
    float inv = svg / l_run[r];
    float* yp = Y + ((size_t)b * Tn + t0 + row0 + r) * Cn + hh * HSn + m16;
    yp[0]  = acc0[r] * inv;
    yp[16] = acc1[r] * inv;
    yp[32] = acc2[r] * inv;
    yp[48] = acc3[r] * inv;
  }
}

// ---------------------------------------------------------------------------
extern "C" void kernel_launch(void* const* d_in, const int* in_sizes, int n_in,
                              void* d_out, int out_size, void* d_ws, size_t ws_size,
                              hipStream_t stream) {
  const float* x           = (const float*)d_in[0];
  const float* W_la        = (const float*)d_in[1];
  const float* leak        = (const float*)d_in[2];
  const float* K_mem       = (const float*)d_in[3];
  const float* V_mem       = (const float*)d_in[4];
  const float* kernel_beta = (const float*)d_in[5];
  const float* value_beta  = (const float*)d_in[6];
  const float* W_cproj     = (const float*)d_in[7];
  float* out = (float*)d_out;

  const int B = 2, T = 2048, C = 768, NH = 12, HS = 64, M = 2688;
  const size_t BTC = (size_t)B * T * C;

  float* u  = (float*)d_ws;       // (B,T,C)
  float* kb = u + BTC;            // (B,NH,T,HS)
  float* y2 = kb + BTC;           // (B,T,C) merged-head attention output

  dim3 gblk(32, 4);

  // 1) u = x @ W_la^T          (each wave: 16x64 stripe)
  gemm_nt_wmma<<<dim3((B * T) / 16, C / 256), gblk, 0, stream>>>(x, W_la, u, B * T, C, C);

  // 2) leaky EMA scan over time -> head-major k
  ema_scan<<<(B * C + 255) / 256, 256, 0, stream>>>(u, leak, kb, B, T, C, NH, HS);

  // 3) attention over persistent memory (fused norm+scale, double-buffered
  //    async K/V staging overlapping the WMMA pipeline)
  pmem_attn<<<(B * NH * (T / 16)) / 4, gblk, 0, stream>>>(kb, K_mem, V_mem,
                                                          kernel_beta, value_beta, y2,
                                                          B, T, NH, M, HS);

  // 4) out = y2 @ W_cproj^T
  gemm_nt_wmma<<<dim3((B * T) / 16, C / 256), gblk, 0, stream>>>(y2, W_cproj, out, B * T, C, C);
}